// SAGE_42949672960221
// MI455X (gfx1250) — compile-verified
//
#include <hip/hip_runtime.h>
#include <hip/hip_bf16.h>

typedef float v2f __attribute__((ext_vector_type(2)));
typedef float v8f __attribute__((ext_vector_type(8)));

#define N_NODES 50000
#define N_EDGES 800000
#define DIM     128
#define CLS     64
#define LAYERS  3
#define EPSVAL  1e-5f
#define ROWPAD  132   // 128 + 4 pad floats: breaks 16-way LDS bank conflicts on A-fragment reads

// ---------------------------------------------------------------------------
// CDNA5 async global->LDS helpers (ASYNCcnt path, ISA 15.18.3 / CDNA5 ISA)
// ---------------------------------------------------------------------------
__device__ __forceinline__ void async_copy_b128(const float* gbase,   // uniform -> SGPR pair
                                                unsigned     gbyteoff, // per-lane byte offset (VGPR)
                                                void*        lds_ptr)  // generic ptr into LDS
{
    // Generic LDS address: low 32 bits are the LDS byte address (aperture rule).
    unsigned lds_addr = (unsigned)(size_t)lds_ptr;
    asm volatile("global_load_async_to_lds_b128 %0, %1, %2"
                 :
                 : "v"(lds_addr), "v"(gbyteoff), "s"(gbase)
                 : "memory");
}

__device__ __forceinline__ void wait_async0() {
#if __has_builtin(__builtin_amdgcn_s_wait_asynccnt)
    __builtin_amdgcn_s_wait_asynccnt(0);
#else
    asm volatile("s_wait_asynccnt 0" ::: "memory");
#endif
}

// ---------------------------------------------------------------------------
// Degree kernels
// ---------------------------------------------------------------------------
__global__ __launch_bounds__(256) void deg_count_kernel(const int* __restrict__ dst,
                                                        float* __restrict__ deg) {
    int e = blockIdx.x * 256 + threadIdx.x;
    if (e < N_EDGES) atomicAdd(&deg[dst[e]], 1.0f);
}

__global__ __launch_bounds__(256) void deg_inv_kernel(float* __restrict__ deg) {
    int i = blockIdx.x * 256 + threadIdx.x;
    if (i < N_NODES) deg[i] = 1.0f / fmaxf(deg[i], 1.0f);
}

// ---------------------------------------------------------------------------
// Edge scatter: one wave32 per edge, float4 per lane covers 128 floats
// ---------------------------------------------------------------------------
__global__ __launch_bounds__(256) void scatter_add_kernel(const float* __restrict__ h,
                                                          const int* __restrict__ src,
                                                          const int* __restrict__ dst,
                                                          float* __restrict__ agg) {
    int wave = (blockIdx.x * 256 + threadIdx.x) >> 5;   // uniform per wave
    int lane = threadIdx.x & 31;
    if (wave >= N_EDGES) return;
    int s = src[wave];
    int d = dst[wave];
    const float4 v = ((const float4*)(h + (size_t)s * DIM))[lane];
    float* ad = agg + (size_t)d * DIM + lane * 4;
    atomicAdd(ad + 0, v.x);
    atomicAdd(ad + 1, v.y);
    atomicAdd(ad + 2, v.z);
    atomicAdd(ad + 3, v.w);
}

// ---------------------------------------------------------------------------
// Fused SAGE layer: h_out = h@Wself + (agg*invdeg)@Wneigh + b + h  [+ReLU+LN]
// Block = 16 nodes x 128 dims, 8 waves, one 16x16 WMMA col-tile per wave.
// Tiles staged via GLOBAL_LOAD_ASYNC_TO_LDS_B128; invdeg scaling folded into
// the A-fragment load (each lane owns a fixed row m = lane&15).
// ---------------------------------------------------------------------------
__global__ __launch_bounds__(256)
void sage_layer_kernel(const float* __restrict__ hin,
                       const float* __restrict__ agg,      // raw sums (unscaled)
                       const float* __restrict__ invdeg,
                       const float* __restrict__ Wself,    // 128x128 row-major
                       const float* __restrict__ Wneigh,   // 128x128 row-major
                       const float* __restrict__ bias,     // 128
                       const float* __restrict__ gamma,    // 128
                       const float* __restrict__ beta,     // 128
                       float* __restrict__ hout,
                       int do_ln)
{
    __shared__ __align__(16) float sH[16][ROWPAD];  // input h tile (also residual)
    __shared__ __align__(16) float sA[16][ROWPAD];  // raw agg tile
    __shared__ __align__(16) float sO[16][ROWPAD];  // post-ReLU staging for LN

    const int rowBase = blockIdx.x * 16;
    const int tid  = threadIdx.x;
    const int wave = tid >> 5;
    const int lane = tid & 31;

    // Prefetch the L2-resident weight matrices (both 64KB, shared by all blocks)
    if (tid < 64) {
        __builtin_prefetch(Wself  + tid * 256, 0, 1);
        __builtin_prefetch(Wneigh + tid * 256, 0, 1);
    }

    // --- async stage of 16x128 h and agg tiles (16B per lane per issue)
    const float* hbase = hin + (size_t)rowBase * DIM;
    const float* abase = agg + (size_t)rowBase * DIM;
    for (int i = tid; i < 16 * 32; i += 256) {
        const int r  = i >> 5;
        const int c4 = i & 31;
        const unsigned goff = (unsigned)((r * DIM + c4 * 4) * 4);
        async_copy_b128(hbase, goff, &sH[r][c4 * 4]);
        async_copy_b128(abase, goff, &sA[r][c4 * 4]);
    }
    wait_async0();
    __syncthreads();

    // --- WMMA: D = sH @ Wself[:,tile] + (idg[m]*sA) @ Wneigh[:,tile]
    const int m  = lane & 15;      // A-fragment row
    const int kh = lane >> 4;      // K half-select (0: K={0,1}, 1: K={2,3})
    const int n  = lane & 15;      // B/C-fragment column
    const int colBase = wave * 16;
    const float idg = invdeg[rowBase + m];   // per-lane row scale for agg

    v8f acc = {0.f, 0.f, 0.f, 0.f, 0.f, 0.f, 0.f, 0.f};

    #pragma unroll
    for (int k0 = 0; k0 < DIM; k0 += 4) {
        const int k = k0 + 2 * kh;
        v2f aH; aH.x = sH[m][k];        aH.y = sH[m][k + 1];
        v2f aA; aA.x = sA[m][k] * idg;  aA.y = sA[m][k + 1] * idg;
        v2f bS; bS.x = Wself [(size_t)k * DIM + colBase + n];
                bS.y = Wself [(size_t)(k + 1) * DIM + colBase + n];
        v2f bN; bN.x = Wneigh[(size_t)k * DIM + colBase + n];
                bN.y = Wneigh[(size_t)(k + 1) * DIM + colBase + n];
        acc = __builtin_amdgcn_wmma_f32_16x16x4_f32(false, aH, false, bS,
                                                    (short)0, acc, false, false);
        acc = __builtin_amdgcn_wmma_f32_16x16x4_f32(false, aA, false, bN,
                                                    (short)0, acc, false, false);
    }

    // --- epilogue: + bias + residual, then ReLU->LDS (LN) or direct store
    const float bcol = bias[colBase + n];
    #pragma unroll
    for (int r = 0; r < 8; ++r) {
        const int mm = r + 8 * kh;                       // C/D layout: M = r + 8*(lane>=16)
        float val = acc[r] + bcol + sH[mm][colBase + n]; // residual from LDS h tile
        if (do_ln) {
            sO[mm][colBase + n] = fmaxf(val, 0.0f);      // ReLU
        } else {
            hout[(size_t)(rowBase + mm) * DIM + colBase + n] = val;
        }
    }

    // --- fused LayerNorm over 128 features (2 rows per wave, wave32 shuffles)
    if (do_ln) {
        __syncthreads();
        #pragma unroll
        for (int j = 0; j < 2; ++j) {
            const int rr = wave * 2 + j;
            const int c  = lane * 4;
            float4 v = *(const float4*)&sO[rr][c];
            float s = v.x + v.y + v.z + v.w;
            #pragma unroll
            for (int off = 16; off > 0; off >>= 1) s += __shfl_xor(s, off, 32);
            const float mu = s * (1.0f / 128.0f);
            float dx = v.x - mu, dy = v.y - mu, dz = v.z - mu, dw = v.w - mu;
            float q = dx * dx + dy * dy + dz * dz + dw * dw;
            #pragma unroll
            for (int off = 16; off > 0; off >>= 1) q += __shfl_xor(q, off, 32);
            const float rstd = rsqrtf(q * (1.0f / 128.0f) + EPSVAL);
            float4 g  = *(const float4*)(gamma + c);
            float4 be = *(const float4*)(beta + c);
            float4 o;
            o.x = g.x * dx * rstd + be.x;
            o.y = g.y * dy * rstd + be.y;
            o.z = g.z * dz * rstd + be.z;
            o.w = g.w * dw * rstd + be.w;
            *(float4*)(hout + (size_t)(rowBase + rr) * DIM + c) = o;
        }
    }
}

// ---------------------------------------------------------------------------
// Output projection: logits = emb(Nx128) @ Wout(128x64) + bout
// Block = 16 nodes, 4 waves x 16-col tiles = 64 classes.
// ---------------------------------------------------------------------------
__global__ __launch_bounds__(128)
void out_gemm_kernel(const float* __restrict__ emb,
                     const float* __restrict__ Wout,
                     const float* __restrict__ bout,
                     float* __restrict__ logits)
{
    __shared__ __align__(16) float sE[16][ROWPAD];
    const int rowBase = blockIdx.x * 16;
    const int tid  = threadIdx.x;
    const int wave = tid >> 5;
    const int lane = tid & 31;

    const float* ebase = emb + (size_t)rowBase * DIM;
    for (int i = tid; i < 16 * 32; i += 128) {
        const int r  = i >> 5;
        const int c4 = i & 31;
        const unsigned goff = (unsigned)((r * DIM + c4 * 4) * 4);
        async_copy_b128(ebase, goff, &sE[r][c4 * 4]);
    }
    wait_async0();
    __syncthreads();

    const int m  = lane & 15;
    const int kh = lane >> 4;
    const int n  = lane & 15;
    const int colBase = wave * 16;

    v8f acc = {0.f, 0.f, 0.f, 0.f, 0.f, 0.f, 0.f, 0.f};
    #pragma unroll
    for (int k0 = 0; k0 < DIM; k0 += 4) {
        const int k = k0 + 2 * kh;
        v2f a; a.x = sE[m][k]; a.y = sE[m][k + 1];
        v2f b; b.x = Wout[(size_t)k * CLS + colBase + n];
               b.y = Wout[(size_t)(k + 1) * CLS + colBase + n];
        acc = __builtin_amdgcn_wmma_f32_16x16x4_f32(false, a, false, b,
                                                    (short)0, acc, false, false);
    }
    const float bb = bout[colBase + n];
    #pragma unroll
    for (int r = 0; r < 8; ++r) {
        const int mm = r + 8 * kh;
        logits[(size_t)(rowBase + mm) * CLS + colBase + n] = acc[r] + bb;
    }
}

// ---------------------------------------------------------------------------
// Host-side orchestration (graph-capture safe: only async ops on `stream`)
// ---------------------------------------------------------------------------
extern "C" void kernel_launch(void* const* d_in, const int* in_sizes, int n_in,
                              void* d_out, int out_size, void* d_ws, size_t ws_size,
                              hipStream_t stream) {
    const float* x      = (const float*)d_in[0];
    const int*   src    = (const int*)  d_in[1];
    const int*   dst    = (const int*)  d_in[2];
    const float* Wself  = (const float*)d_in[3];   // [L,128,128]
    const float* Wneigh = (const float*)d_in[4];   // [L,128,128]
    const float* bconv  = (const float*)d_in[5];   // [L,128]
    const float* gamma  = (const float*)d_in[6];   // [L-1,128]
    const float* beta   = (const float*)d_in[7];   // [L-1,128]
    const float* Wout   = (const float*)d_in[8];   // [128,64]
    const float* bout   = (const float*)d_in[9];   // [64]

    float* logits = (float*)d_out;                                   // N x 64
    float* emb    = (float*)d_out + (size_t)N_NODES * CLS;           // N x 128

    // workspace partition
    char*  ws     = (char*)d_ws;
    size_t off    = 0;
    float* invdeg = (float*)(ws + off); off += ((size_t)N_NODES * 4 + 255) & ~(size_t)255;
    float* agg    = (float*)(ws + off); off += (size_t)N_NODES * DIM * 4;
    float* h1     = (float*)(ws + off); off += (size_t)N_NODES * DIM * 4;
    float* h2     = (float*)(ws + off); off += (size_t)N_NODES * DIM * 4;

    // --- degree / inverse degree
    hipMemsetAsync(invdeg, 0, (size_t)N_NODES * 4, stream);
    deg_count_kernel<<<(N_EDGES + 255) / 256, 256, 0, stream>>>(dst, invdeg);
    deg_inv_kernel<<<(N_NODES + 255) / 256, 256, 0, stream>>>(invdeg);

    const int nRowTiles   = N_NODES / 16;               // 3125, exact
    const int scatterBlks = (N_EDGES * 32 + 255) / 256; // one wave per edge

    const float* hcur = x;
    float* bufs[LAYERS] = {h1, h2, emb};
    for (int l = 0; l < LAYERS; ++l) {
        hipMemsetAsync(agg, 0, (size_t)N_NODES * DIM * 4, stream);
        scatter_add_kernel<<<scatterBlks, 256, 0, stream>>>(hcur, src, dst, agg);
        const int do_ln = (l != LAYERS - 1);
        sage_layer_kernel<<<nRowTiles, 256, 0, stream>>>(
            hcur, agg, invdeg,
            Wself  + (size_t)l * DIM * DIM,
            Wneigh + (size_t)l * DIM * DIM,
            bconv  + (size_t)l * DIM,
            do_ln ? gamma + (size_t)l * DIM : gamma,
            do_ln ? beta  + (size_t)l * DIM : beta,
            bufs[l], do_ln);
        hcur = bufs[l];
    }

    out_gemm_kernel<<<nRowTiles, 128, 0, stream>>>(emb, Wout, bout, logits);
}